// NNConvPair_77738908058168
// MI455X (gfx1250) — compile-verified
//
#include <hip/hip_runtime.h>
#include <hip/hip_bf16.h>

// ---------------------------------------------------------------------------
// NNConvPair on MI455X (gfx1250, wave32, WMMA + TDM).
// Dominant FLOPs: per-edge  mid[16] @ w2[16, in_ch*16]  -> done with
// V_WMMA_F32_16X16X4_F32 on tiles of 16 edges, fully fused so the per-edge
// weight tensor W[E, in_ch, 16] (410 MB) is never materialized in HBM.
// w2 weight staging into LDS uses the Tensor Data Mover when available.
// ---------------------------------------------------------------------------

typedef float v2f __attribute__((ext_vector_type(2)));
typedef float v8f __attribute__((ext_vector_type(8)));
typedef unsigned int v4u __attribute__((ext_vector_type(4)));
typedef int v4i __attribute__((ext_vector_type(4)));
typedef int v8i __attribute__((ext_vector_type(8)));

#if __has_builtin(__builtin_amdgcn_tensor_load_to_lds) && \
    __has_builtin(__builtin_amdgcn_s_wait_tensorcnt)
#define USE_TDM 1
#endif

__device__ __forceinline__ v8f wmma4(v2f a, v2f b, v8f c) {
  // D(16x16,f32) = A(16x4,f32) * B(4x16,f32) + C
  return __builtin_amdgcn_wmma_f32_16x16x4_f32(
      /*neg_a=*/false, a, /*neg_b=*/false, b,
      /*c_mod=*/(short)0, c, /*reuse_a=*/false, /*reuse_b=*/false);
}

// monotonic float<->uint encoding for atomicMax-based segment max
__device__ __forceinline__ unsigned enc_f32(float x) {
  unsigned u = __float_as_uint(x);
  return (u & 0x80000000u) ? ~u : (u | 0x80000000u);
}
__device__ __forceinline__ float dec_f32(unsigned v) {
  unsigned u = (v & 0x80000000u) ? (v & 0x7FFFFFFFu) : ~v;
  return __uint_as_float(u);
}

#ifdef USE_TDM
// TDM copy of `nelem` contiguous f32 from global `gsrc` into LDS offset
// `lds_byte_addr` (1-row 1-D tile; D# per CDNA5 ISA ch.8). Issued once per
// wave that calls it; EXEC is ignored by TENSOR_LOAD_TO_LDS.
__device__ __forceinline__ void tdm_copy_f32(const float* gsrc,
                                             unsigned lds_byte_addr,
                                             unsigned nelem /* <= 65535 */) {
  const unsigned long long ga = (unsigned long long)(const void*)gsrc;
  v4u g0;
  g0.x = 1u;                                        // count=1, load, no gather
  g0.y = lds_byte_addr;                             // lds_addr [63:32]
  g0.z = (unsigned)ga;                              // global_addr [95:64]
  g0.w = (unsigned)((ga >> 32) & 0x1FFFFFFu) | 0x80000000u;  // addr hi | type=2
  v8i g1;
  g1[0] = 0x00020000;                 // workgroup_mask=0, data_size=4B
  g1[1] = (int)(nelem << 16);         // tensor_dim0[15:0] at [63:48]
  g1[2] = (int)(1u << 16);            // tensor_dim0[31:16]=0, tensor_dim1=1
  g1[3] = (int)(nelem << 16);         // tile_dim0 at [127:112]
  g1[4] = 1;                          // tile_dim1=1, tile_dim2=0
  g1[5] = (int)nelem;                 // tensor_dim0_stride[31:0]
  g1[6] = (int)(nelem << 16);         // stride0[47:32]=0, stride1[15:0]
  g1[7] = 0;                          // stride1[47:16]=0
  v4i g2 = {0, 0, 0, 0};              // tensor_dim2/3=0, no iterate
  v4i g3 = {0, 0, 0, 0};
#if defined(__clang_major__) && __clang_major__ >= 23
  v8i g1x = {0, 0, 0, 0, 0, 0, 0, 0};
  __builtin_amdgcn_tensor_load_to_lds(g0, g1, g2, g3, g1x, 0);
#else
  __builtin_amdgcn_tensor_load_to_lds(g0, g1, g2, g3, 0);
#endif
}
#endif  // USE_TDM

// ---------------------------------------------------------------------------
// Fused NNConv edge kernel.
//   x   : [N, IN_CH]  (relu applied on load -> conv input is relu(x))
//   ea  : [E, 8]
//   out : [N, 16] pre-initialized with relu(x)@root + bias; we atomicAdd msg.
// One wave handles 16 edges: 4 waves / block of 128 threads.
// ---------------------------------------------------------------------------
template <int IN_CH>
__global__ __launch_bounds__(128) void edge_conv_kernel(
    const float* __restrict__ x, const float* __restrict__ ea,
    const int* __restrict__ src, const int* __restrict__ dst,
    const float* __restrict__ w1, const float* __restrict__ b1,
    const float* __restrict__ w2, const float* __restrict__ b2,
    float* __restrict__ out, int E) {
  constexpr int WCOLS = IN_CH * 16;  // w2 is [16, WCOLS]
  constexpr int Q = IN_CH / 4;       // float4s per x row

  __shared__ float s_w1[8 * 16];
  __shared__ float s_b1[16];
  __shared__ __align__(16) float s_w2[16 * WCOLS];
  __shared__ float s_b2[WCOLS];
  __shared__ float s_mid[4][16 * 16];
  __shared__ __align__(16) float s_x[4][16 * IN_CH];
  __shared__ int s_srcidx[4][16];
  __shared__ int s_dstidx[4][16];

  const int t = threadIdx.x;
  const int wave = t >> 5;
  const int lane = t & 31;
  const int n = lane & 15;   // N-index (also A-row index for this lane)
  const int hk = lane >> 4;  // lane-half selector for K split
  const int e0 = (blockIdx.x * 4 + wave) * 16;

  // ---- phase 1: stage weights (+ per-tile edge indices) into LDS ----------
#ifdef USE_TDM
  if (t == 0) {
    tdm_copy_f32(w2, (unsigned)(size_t)(const void*)&s_w2[0], 16u * WCOLS);
  }
#else
  for (int i = t; i < 16 * WCOLS; i += 128) s_w2[i] = w2[i];
#endif
  for (int i = t; i < 8 * 16; i += 128) s_w1[i] = w1[i];
  if (t < 16) s_b1[t] = b1[t];
  for (int i = t; i < WCOLS; i += 128) s_b2[i] = b2[i];
  {
    const int ee = e0 + n;
    const bool vld = (ee < E);
    if (lane < 16) s_srcidx[wave][n] = vld ? src[ee] : -1;
    else           s_dstidx[wave][n] = vld ? dst[ee] : -1;
  }
#ifdef USE_TDM
  __builtin_amdgcn_s_wait_tensorcnt(0);
#endif
  __syncthreads();

  // ---- phase 2a: mid = relu(ea @ w1 + b1)   [16 edges x 16], K = 8 -------
  v2f a0, a1;
  {
    const int er = e0 + n;
    if (er < E) {
      const float2* row2 = (const float2*)(ea + (size_t)er * 8);
      const float2 u0 = row2[hk];      // k = 2hk, 2hk+1
      const float2 u1 = row2[2 + hk];  // k = 4+2hk, 5+2hk
      a0.x = u0.x; a0.y = u0.y;
      a1.x = u1.x; a1.y = u1.y;
    } else {
      a0.x = a0.y = a1.x = a1.y = 0.f;
    }
  }
  v2f wb0, wb1;
  wb0.x = s_w1[(2 * hk) * 16 + n];     wb0.y = s_w1[(2 * hk + 1) * 16 + n];
  wb1.x = s_w1[(4 + 2 * hk) * 16 + n]; wb1.y = s_w1[(5 + 2 * hk) * 16 + n];
  const float bc1 = s_b1[n];
  v8f mid = {bc1, bc1, bc1, bc1, bc1, bc1, bc1, bc1};
  mid = wmma4(a0, wb0, mid);
  mid = wmma4(a1, wb1, mid);
#pragma unroll
  for (int r = 0; r < 8; ++r) mid[r] = fmaxf(mid[r], 0.f);

  // store mid (C layout -> row major [edge][k]) into per-wave LDS tile
  float* sm = s_mid[wave];
#pragma unroll
  for (int r = 0; r < 8; ++r) sm[(r + 8 * hk) * 16 + n] = mid[r];

  // ---- phase 2b: gather relu(x[src]) tile [16 x IN_CH] via b128 loads ----
  float* sx = s_x[wave];
  {
    const float4* __restrict__ x4 = (const float4*)x;
    float4* sx4 = (float4*)sx;
#pragma unroll
    for (int itr = 0; itr < (16 * Q) / 32; ++itr) {
      const int idx = lane + 32 * itr;
      const int mm = idx / Q;
      const int q = idx % Q;
      const int node = s_srcidx[wave][mm];
      float4 v = {0.f, 0.f, 0.f, 0.f};
      if (node >= 0) {
        const float4 u = x4[(size_t)node * Q + q];
        v.x = fmaxf(u.x, 0.f); v.y = fmaxf(u.y, 0.f);
        v.z = fmaxf(u.z, 0.f); v.w = fmaxf(u.w, 0.f);
      }
      sx4[idx] = v;
    }
  }
  __syncthreads();

  // re-load mid in A layout (K = 16 -> 4 K-steps)
  const float* amr = sm + n * 16;
  v2f A0, A1, A2, A3;
  A0.x = amr[2 * hk];      A0.y = amr[2 * hk + 1];
  A1.x = amr[4 + 2 * hk];  A1.y = amr[5 + 2 * hk];
  A2.x = amr[8 + 2 * hk];  A2.y = amr[9 + 2 * hk];
  A3.x = amr[12 + 2 * hk]; A3.y = amr[13 + 2 * hk];

  // ---- phase 3: for each input channel i:
  //   T_i = mid @ w2[:, 16i:16i+16] + b2[16i:16i+16]   (4 WMMAs, K=16)
  //   msg[e, :] += x_src[e, i] * T_i[e, :]
  v8f msg = {0.f, 0.f, 0.f, 0.f, 0.f, 0.f, 0.f, 0.f};
#pragma unroll 2
  for (int i = 0; i < IN_CH; ++i) {
    const int col = i * 16 + n;
    const float bc2 = s_b2[col];
    v2f B0, B1, B2, B3;
    B0.x = s_w2[(2 * hk) * WCOLS + col];      B0.y = s_w2[(2 * hk + 1) * WCOLS + col];
    B1.x = s_w2[(4 + 2 * hk) * WCOLS + col];  B1.y = s_w2[(5 + 2 * hk) * WCOLS + col];
    B2.x = s_w2[(8 + 2 * hk) * WCOLS + col];  B2.y = s_w2[(9 + 2 * hk) * WCOLS + col];
    B3.x = s_w2[(12 + 2 * hk) * WCOLS + col]; B3.y = s_w2[(13 + 2 * hk) * WCOLS + col];
    v8f T = {bc2, bc2, bc2, bc2, bc2, bc2, bc2, bc2};
    T = wmma4(A0, B0, T);
    T = wmma4(A1, B1, T);
    T = wmma4(A2, B2, T);
    T = wmma4(A3, B3, T);
#pragma unroll
    for (int r = 0; r < 8; ++r) {
      const float xv = sx[(r + 8 * hk) * IN_CH + i];
      msg[r] = fmaf(xv, T[r], msg[r]);
    }
  }

  // ---- scatter-add messages to destination nodes -------------------------
#pragma unroll
  for (int r = 0; r < 8; ++r) {
    const int d = s_dstidx[wave][r + 8 * hk];
    if (d >= 0) {
      atomicAdd(&out[(size_t)d * 16 + n], msg[r]);
    }
  }
}

// ---------------------------------------------------------------------------
// out[n, o] = sum_c relu(x[n,c]) * root[c,o] + bias[o]
// ---------------------------------------------------------------------------
template <int IN_CH>
__global__ void root_bias_kernel(const float* __restrict__ x,
                                 const float* __restrict__ root,
                                 const float* __restrict__ bias,
                                 float* __restrict__ out, int N) {
  const int idx = blockIdx.x * blockDim.x + threadIdx.x;
  if (idx >= N * 16) return;
  const int node = idx >> 4;
  const int o = idx & 15;
  float acc = bias[o];
#pragma unroll
  for (int c = 0; c < IN_CH; ++c)
    acc = fmaf(fmaxf(x[(size_t)node * IN_CH + c], 0.f), root[c * 16 + o], acc);
  out[idx] = acc;
}

// ---------------------------------------------------------------------------
// attention pool (3 stages) + mean pool + head
// ---------------------------------------------------------------------------
__global__ void gate_kernel(const float* __restrict__ x,
                            const int* __restrict__ batch,
                            const float* __restrict__ gw1,
                            const float* __restrict__ gb1,
                            const float* __restrict__ gw2,
                            const float* __restrict__ gb2,
                            float* __restrict__ gate,
                            unsigned* __restrict__ menc, int N) {
  const int nid = blockIdx.x * blockDim.x + threadIdx.x;
  if (nid >= N) return;
  const float* xr = x + (size_t)nid * 32;
  float g = gb2[0];
  for (int j = 0; j < 32; ++j) {
    float s = gb1[j];
    for (int c = 0; c < 32; ++c) s = fmaf(xr[c], gw1[c * 32 + j], s);
    g = fmaf(fmaxf(s, 0.f), gw2[j], g);
  }
  gate[nid] = g;
  atomicMax(&menc[batch[nid]], enc_f32(g));
}

__global__ void expden_kernel(const float* __restrict__ gate,
                              const int* __restrict__ batch,
                              const unsigned* __restrict__ menc,
                              float* __restrict__ ebuf,
                              float* __restrict__ denom, int N) {
  const int nid = blockIdx.x * blockDim.x + threadIdx.x;
  if (nid >= N) return;
  const int b = batch[nid];
  const float e = expf(gate[nid] - dec_f32(menc[b]));
  ebuf[nid] = e;
  atomicAdd(&denom[b], e);
}

__global__ void attsum_kernel(const float* __restrict__ x,
                              const int* __restrict__ batch,
                              const float* __restrict__ ebuf,
                              const float* __restrict__ denom,
                              float* __restrict__ pooled, int N) {
  const int idx = blockIdx.x * blockDim.x + threadIdx.x;
  if (idx >= N * 32) return;
  const int nid = idx >> 5;
  const int c = idx & 31;
  const int b = batch[nid];
  const float a = ebuf[nid] / (denom[b] + 1e-16f);
  atomicAdd(&pooled[(size_t)b * 32 + c], a * x[(size_t)nid * 32 + c]);
}

__global__ void meansum_kernel(const float* __restrict__ h,
                               const int* __restrict__ batch,
                               float* __restrict__ sums,
                               float* __restrict__ cnt, int N) {
  const int idx = blockIdx.x * blockDim.x + threadIdx.x;
  if (idx >= N * 16) return;
  const int nid = idx >> 4;
  const int o = idx & 15;
  const int b = batch[nid];
  atomicAdd(&sums[(size_t)b * 16 + o], h[idx]);
  if (o == 0) atomicAdd(&cnt[b], 1.0f);
}

__global__ void head_kernel(const float* __restrict__ msum_p,
                            const float* __restrict__ msum_d,
                            const float* __restrict__ cnt_p,
                            const float* __restrict__ cnt_d,
                            const float* __restrict__ att_p,
                            const float* __restrict__ att_d,
                            const float* __restrict__ l1w,
                            const float* __restrict__ l1b,
                            const float* __restrict__ l2w,
                            const float* __restrict__ l2b,
                            float* __restrict__ out, int B) {
  const int b = blockIdx.x * blockDim.x + threadIdx.x;
  if (b >= B) return;
  float f[96];
  const float cp = fmaxf(cnt_p[b], 1.f), cd = fmaxf(cnt_d[b], 1.f);
  for (int o = 0; o < 16; ++o) f[o] = msum_p[b * 16 + o] / cp;
  for (int o = 0; o < 16; ++o) f[16 + o] = msum_d[b * 16 + o] / cd;
  for (int c = 0; c < 32; ++c) f[32 + c] = att_p[b * 32 + c];
  for (int c = 0; c < 32; ++c) f[64 + c] = att_d[b * 32 + c];
  float y = l2b[0];
  for (int j = 0; j < 8; ++j) {
    float s = l1b[j];
    for (int k = 0; k < 96; ++k) s = fmaf(f[k], l1w[k * 8 + j], s);
    y = fmaf(s, l2w[j], y);
  }
  out[b] = y;
}

__global__ void zero_kernel(float* __restrict__ p, int n) {
  const int i = blockIdx.x * blockDim.x + threadIdx.x;
  if (i < n) p[i] = 0.f;
}

// ---------------------------------------------------------------------------
extern "C" void kernel_launch(void* const* d_in, const int* in_sizes, int n_in,
                              void* d_out, int out_size, void* d_ws,
                              size_t ws_size, hipStream_t stream) {
  (void)n_in; (void)ws_size;
  const int N = in_sizes[6];      // batch_p
  const int E = in_sizes[2] / 8;  // edge_attr_p [E,8]
  const int B = out_size;         // output is [B,1]

  const float* x_p = (const float*)d_in[0];
  const float* x_d = (const float*)d_in[1];
  const float* ea_p = (const float*)d_in[2];
  const float* ea_d = (const float*)d_in[3];
  const int* ei_p = (const int*)d_in[4];
  const int* ei_d = (const int*)d_in[5];
  const int* batch_p = (const int*)d_in[6];
  const int* batch_d = (const int*)d_in[7];

  auto F = [&](int i) { return (const float*)d_in[i]; };
  // nested-dict flatten order sniffing: insertion {w1,b1,w2,b2,root,bias}
  // vs alphabetical {b1,b2,bias,root,w1,w2}. conv1.w1 has 8*16=128 elements.
  const bool convIns = (in_sizes[8] == 128);
  const int o_w1 = convIns ? 0 : 4, o_b1 = convIns ? 1 : 0,
            o_w2 = convIns ? 2 : 5, o_b2 = convIns ? 3 : 1,
            o_rt = convIns ? 4 : 3, o_bs = convIns ? 5 : 2;
  // params_d: conv1 @8, conv2 @14 ; params_p: conv1 @20, conv2 @26
  const float *d1_w1 = F(8 + o_w1), *d1_b1 = F(8 + o_b1), *d1_w2 = F(8 + o_w2),
              *d1_b2 = F(8 + o_b2), *d1_rt = F(8 + o_rt), *d1_bs = F(8 + o_bs);
  const float *d2_w1 = F(14 + o_w1), *d2_b1 = F(14 + o_b1), *d2_w2 = F(14 + o_w2),
              *d2_b2 = F(14 + o_b2), *d2_rt = F(14 + o_rt), *d2_bs = F(14 + o_bs);
  const float *p1_w1 = F(20 + o_w1), *p1_b1 = F(20 + o_b1), *p1_w2 = F(20 + o_w2),
              *p1_b2 = F(20 + o_b2), *p1_rt = F(20 + o_rt), *p1_bs = F(20 + o_bs);
  const float *p2_w1 = F(26 + o_w1), *p2_b1 = F(26 + o_b1), *p2_w2 = F(26 + o_w2),
              *p2_b2 = F(26 + o_b2), *p2_rt = F(26 + o_rt), *p2_bs = F(26 + o_bs);
  const bool poolIns = (in_sizes[32] == 32 * 32);  // gw1 first?
  const float* gw1 = F(32 + (poolIns ? 0 : 2));
  const float* gb1 = F(32 + (poolIns ? 1 : 0));
  const float* gw2 = F(32 + (poolIns ? 2 : 3));
  const float* gb2 = F(32 + (poolIns ? 3 : 1));
  const bool headIns = (in_sizes[36] == 96 * 8);  // l1w first?
  const float* l1w = F(36 + (headIns ? 0 : 1));
  const float* l1b = F(36 + (headIns ? 1 : 0));
  const float* l2w = F(36 + (headIns ? 2 : 3));
  const float* l2b = F(36 + (headIns ? 3 : 2));

  // ---- workspace carving (zeroed region first) ----------------------------
  float* ws = (float*)d_ws;
  size_t off = 0;
  auto alloc = [&](size_t cnt) { float* p = ws + off; off += cnt; return p; };
  unsigned* menc_p = (unsigned*)alloc(B);
  unsigned* menc_d = (unsigned*)alloc(B);
  float* denom_p = alloc(B);
  float* denom_d = alloc(B);
  float* att_p = alloc((size_t)B * 32);
  float* att_d = alloc((size_t)B * 32);
  float* msum_p = alloc((size_t)B * 16);
  float* msum_d = alloc((size_t)B * 16);
  float* cnt_p = alloc(B);
  float* cnt_d = alloc(B);
  const int zcount = (int)off;
  float* gate_p = alloc(N);
  float* gate_d = alloc(N);
  float* e_p = alloc(N);
  float* e_d = alloc(N);
  float* h1_p = alloc((size_t)N * 16);
  float* h1_d = alloc((size_t)N * 16);
  float* h2_p = alloc((size_t)N * 16);
  float* h2_d = alloc((size_t)N * 16);

  const int* src_p = ei_p;
  const int* dst_p = ei_p + E;
  const int* src_d = ei_d;
  const int* dst_d = ei_d + E;

  zero_kernel<<<(zcount + 255) / 256, 256, 0, stream>>>(ws, zcount);

  // attention pooling (on raw x)
  const int gN = (N + 127) / 128;
  gate_kernel<<<gN, 128, 0, stream>>>(x_p, batch_p, gw1, gb1, gw2, gb2, gate_p, menc_p, N);
  gate_kernel<<<gN, 128, 0, stream>>>(x_d, batch_d, gw1, gb1, gw2, gb2, gate_d, menc_d, N);
  expden_kernel<<<gN, 128, 0, stream>>>(gate_p, batch_p, menc_p, e_p, denom_p, N);
  expden_kernel<<<gN, 128, 0, stream>>>(gate_d, batch_d, menc_d, e_d, denom_d, N);
  const int aN = (N * 32 + 255) / 256;
  attsum_kernel<<<aN, 256, 0, stream>>>(x_p, batch_p, e_p, denom_p, att_p, N);
  attsum_kernel<<<aN, 256, 0, stream>>>(x_d, batch_d, e_d, denom_d, att_d, N);

  // conv1 (in=32 -> 16); params_d on p-graph, params_p on d-graph
  const int rN = (N * 16 + 255) / 256;
  const int eGrid = (E + 63) / 64;
  root_bias_kernel<32><<<rN, 256, 0, stream>>>(x_p, d1_rt, d1_bs, h1_p, N);
  root_bias_kernel<32><<<rN, 256, 0, stream>>>(x_d, p1_rt, p1_bs, h1_d, N);
  edge_conv_kernel<32><<<eGrid, 128, 0, stream>>>(x_p, ea_p, src_p, dst_p,
                                                  d1_w1, d1_b1, d1_w2, d1_b2, h1_p, E);
  edge_conv_kernel<32><<<eGrid, 128, 0, stream>>>(x_d, ea_d, src_d, dst_d,
                                                  p1_w1, p1_b1, p1_w2, p1_b2, h1_d, E);

  // conv2 (in=16 -> 16), input relu(h1) applied on load
  root_bias_kernel<16><<<rN, 256, 0, stream>>>(h1_p, d2_rt, d2_bs, h2_p, N);
  root_bias_kernel<16><<<rN, 256, 0, stream>>>(h1_d, p2_rt, p2_bs, h2_d, N);
  edge_conv_kernel<16><<<eGrid, 128, 0, stream>>>(h1_p, ea_p, src_p, dst_p,
                                                  d2_w1, d2_b1, d2_w2, d2_b2, h2_p, E);
  edge_conv_kernel<16><<<eGrid, 128, 0, stream>>>(h1_d, ea_d, src_d, dst_d,
                                                  p2_w1, p2_b1, p2_w2, p2_b2, h2_d, E);

  // mean pooling + head
  meansum_kernel<<<rN, 256, 0, stream>>>(h2_p, batch_p, msum_p, cnt_p, N);
  meansum_kernel<<<rN, 256, 0, stream>>>(h2_d, batch_d, msum_d, cnt_d, N);
  head_kernel<<<(B + 63) / 64, 64, 0, stream>>>(msum_p, msum_d, cnt_p, cnt_d,
                                                att_p, att_d, l1w, l1b, l2w, l2b,
                                                (float*)d_out, B);
}